// FilterAugment_27599459844839
// MI455X (gfx1250) — compile-verified
//
#include <hip/hip_runtime.h>
#include <cstdint>

// FilterAugment: out[b,0,f,t] = features[b,0,f,t] * 10^(gain_db(b,f)/20)
// B=64, F=256, T=2048. Pure streaming kernel, HBM-bound (~268 MB total,
// ~11.5us floor at 23.3 TB/s). CDNA5 path: async global->LDS staging
// (ASYNCcnt) issued *before* the scalar gain prologue to overlap DMA with
// KMcnt waits, then ds_load_b128 -> v_mul -> non-temporal b128 stores
// (write-once stream; don't churn the 192MB L2).

#ifndef FA_B
#define FA_B 64
#define FA_F 256
#define FA_T 2048
#endif

typedef float v4f __attribute__((ext_vector_type(4)));

__global__ __launch_bounds__(256) void filter_augment_kernel(
    const float* __restrict__ features,      // (B,1,F,T)
    const float* __restrict__ band_factors,  // (B,5)
    const int*   __restrict__ bnd,           // (5,)
    float*       __restrict__ out)           // (B,1,F,T)
{
    __shared__ float smem[FA_T];             // 8 KB: one row

    const int tid = threadIdx.x;
    const int f   = blockIdx.x & (FA_F - 1);
    const int b   = blockIdx.x >> 8;         // FA_F == 256

    const size_t row = (size_t)blockIdx.x * (size_t)FA_T;
    const float* src = features + row;
    float*       dst = out + row;

    // ---- issue async row staging FIRST: 256 thr x 16 B = 4096 B/issue,
    // two issues cover the 8192 B row. INST_OFFSET is added to BOTH the
    // global and LDS addresses, so the second issue reuses the same
    // LDS-address VGPR with offset:4096. NT hint: read-once stream.
    unsigned lds_off = (unsigned)(uintptr_t)(&smem[tid * 4]); // raw LDS byte offset
    unsigned voff    = (unsigned)(tid * 16);                  // byte offset in row
    unsigned long long sbase = (unsigned long long)(uintptr_t)src;

    asm volatile(
        "global_load_async_to_lds_b128 %0, %1, %2 th:TH_LOAD_NT\n\t"
        "global_load_async_to_lds_b128 %0, %1, %2 offset:4096 th:TH_LOAD_NT"
        :
        : "v"(lds_off), "v"(voff), "s"(sbase)
        : "memory");

    // ---- per-row gain (scalar; overlaps with the async DMA above) ----
    // idx = clip(searchsorted(bnd, f, right) - 1, 0, 3)
    int pos = 0;
#pragma unroll
    for (int k = 0; k < 5; ++k) pos += (f >= bnd[k]) ? 1 : 0;
    int idx = pos - 1;
    idx = idx < 0 ? 0 : (idx > 3 ? 3 : idx);

    const int   lo    = bnd[idx];
    const int   width = bnd[idx + 1] - lo;
    const int   wm1   = (width - 1) > 1 ? (width - 1) : 1;   // max(width-1, 1)
    const float t     = (float)(f - lo) / (float)wm1;
    const float g0    = band_factors[b * 5 + idx];
    const float g1    = band_factors[b * 5 + idx + 1];
    const float gain  = g0 + (g1 - g0) * t;
    // 10^(gain/20) == 2^(gain * log2(10)/20)
    const float filt  = exp2f(gain * 0.16609640474436812f);

    // ---- now wait for the DMA and consume ----
    asm volatile("s_wait_asynccnt 0" ::: "memory");
    __syncthreads();  // waves only read lanes they staged, but be safe

    v4f a = *reinterpret_cast<const v4f*>(&smem[tid * 4]);
    v4f c = *reinterpret_cast<const v4f*>(&smem[tid * 4 + 1024]);
    a *= filt;
    c *= filt;
    __builtin_nontemporal_store(a, reinterpret_cast<v4f*>(dst + tid * 4));
    __builtin_nontemporal_store(c, reinterpret_cast<v4f*>(dst + tid * 4 + 1024));
}

extern "C" void kernel_launch(void* const* d_in, const int* in_sizes, int n_in,
                              void* d_out, int out_size, void* d_ws, size_t ws_size,
                              hipStream_t stream) {
    (void)n_in; (void)d_ws; (void)ws_size; (void)in_sizes;
    const float* features     = (const float*)d_in[0];  // B*1*F*T fp32
    const float* band_factors = (const float*)d_in[1];  // B*5 fp32
    const int*   bnd          = (const int*)d_in[2];    // 5 int32
    float*       out          = (float*)d_out;

    const int n_rows = out_size / FA_T;                 // B*F = 16384
    filter_augment_kernel<<<dim3(n_rows), dim3(256), 0, stream>>>(
        features, band_factors, bnd, out);
}